// Where2comm_23476291240409
// MI455X (gfx1250) — compile-verified
//
#include <hip/hip_runtime.h>
#include <cstdint>

// AttenFusion (Where2comm ATTEN) — ego-row attention over N=5 CAVs per pixel.
// B=16, N=5, C=64, H=100, W=252 -> 403200 independent pixels.
// Memory-bound (~1 FLOP/byte): single-HBM-read design using CDNA5 async
// global->LDS DMA (ASYNCcnt) so score pass and context pass both read LDS.

#define B_   16
#define N_   5
#define C_   64
#define HW_  25200          // 100*252
#define BLK  64             // threads per block = 2 wave32
#define MC_  (N_ * C_)      // 320 staged floats per location

__global__ __launch_bounds__(BLK)
void atten_fuse_kernel(const float* __restrict__ x, float* __restrict__ out)
{
    // 320 * 64 * 4B = 80 KB -> 4 workgroups per WGP (320 KB LDS)
    __shared__ float smem[MC_ * BLK];

    const int tid = threadIdx.x;
    const int g   = blockIdx.x * BLK + tid;      // 6300*64 == 403200 exactly
    const int b   = g / HW_;
    const int p   = g - b * HW_;

    // x[b, m, c, p] : flattening (m,c) -> mc gives element stride HW_.
    const float* gq = x + (size_t)b * (N_ * C_ * HW_) + p;

    // Workgroup-relative LDS byte address of this lane's slot 0.
    // (On gfx1250 the low 32 bits of a generic pointer into LDS are the
    //  LDS byte offset; async-to-LDS VDST wants exactly that.)
    uint32_t lq = (uint32_t)(uintptr_t)(&smem[tid]);

    // ---- Stage: 320 async DMA transfers per wave, each 32 lanes x 4B = 128B
    //      of contiguous HBM (lanes cover consecutive p). Data lands in LDS
    //      without passing through VGPRs; tracked by ASYNCcnt.
    #pragma unroll 4
    for (int mc = 0; mc < MC_; ++mc) {
        asm volatile("global_load_async_to_lds_b32 %0, %1, off"
                     :: "v"(lq), "v"(gq)
                     : "memory");
        gq += HW_;          // next (m,c)
        lq += BLK * 4;      // next LDS slot row
    }
    asm volatile("s_wait_asynccnt 0x0" ::: "memory");

    // ---- Pass A: scores s_m = <x0, xm> (only ego row of the 5x5 needed).
    float s0 = 0.f, s1 = 0.f, s2 = 0.f, s3 = 0.f, s4 = 0.f;
    #pragma unroll 4
    for (int c = 0; c < C_; ++c) {
        const float f0 = smem[(0 * C_ + c) * BLK + tid];
        const float f1 = smem[(1 * C_ + c) * BLK + tid];
        const float f2 = smem[(2 * C_ + c) * BLK + tid];
        const float f3 = smem[(3 * C_ + c) * BLK + tid];
        const float f4 = smem[(4 * C_ + c) * BLK + tid];
        s0 = fmaf(f0, f0, s0);
        s1 = fmaf(f0, f1, s1);
        s2 = fmaf(f0, f2, s2);
        s3 = fmaf(f0, f3, s3);
        s4 = fmaf(f0, f4, s4);
    }

    // ---- Softmax over the 5 CAVs (scale = 1/sqrt(64) = 0.125).
    const float scale = 0.125f;
    s0 *= scale; s1 *= scale; s2 *= scale; s3 *= scale; s4 *= scale;
    const float mx = fmaxf(fmaxf(fmaxf(s0, s1), fmaxf(s2, s3)), s4);
    float w0 = __expf(s0 - mx);
    float w1 = __expf(s1 - mx);
    float w2 = __expf(s2 - mx);
    float w3 = __expf(s3 - mx);
    float w4 = __expf(s4 - mx);
    const float inv = 1.0f / (w0 + w1 + w2 + w3 + w4);
    w0 *= inv; w1 *= inv; w2 *= inv; w3 *= inv; w4 *= inv;

    // ---- Pass B: ego context = sum_m w_m * x_m, straight back out (coalesced).
    float* op = out + (size_t)b * (C_ * HW_) + p;
    #pragma unroll 4
    for (int c = 0; c < C_; ++c) {
        const float f0 = smem[(0 * C_ + c) * BLK + tid];
        const float f1 = smem[(1 * C_ + c) * BLK + tid];
        const float f2 = smem[(2 * C_ + c) * BLK + tid];
        const float f3 = smem[(3 * C_ + c) * BLK + tid];
        const float f4 = smem[(4 * C_ + c) * BLK + tid];
        float r = w0 * f0;
        r = fmaf(w1, f1, r);
        r = fmaf(w2, f2, r);
        r = fmaf(w3, f3, r);
        r = fmaf(w4, f4, r);
        op[(size_t)c * HW_] = r;
    }
}

extern "C" void kernel_launch(void* const* d_in, const int* in_sizes, int n_in,
                              void* d_out, int out_size, void* d_ws, size_t ws_size,
                              hipStream_t stream)
{
    const float* x  = (const float*)d_in[0];
    float* out      = (float*)d_out;
    (void)in_sizes; (void)n_in; (void)out_size; (void)d_ws; (void)ws_size;

    const int total  = B_ * HW_;          // 403200
    const int blocks = total / BLK;       // 6300, exact
    atten_fuse_kernel<<<dim3(blocks), dim3(BLK), 0, stream>>>(x, out);
}